// MLoss_38285338476689
// MI455X (gfx1250) — compile-verified
//
#include <hip/hip_runtime.h>

// ---------------------------------------------------------------------------
// MLoss reduction for MI455X (gfx1250, wave32).
// Pass 1: bandwidth-bound streaming reduce (4 cells/thread via aligned 16B
//         vector loads, non-temporal since 221MB streams once through 192MB L2),
//         wave32 shfl_xor reduce + LDS cross-wave reduce -> 4 partials/block.
// Pass 2: single-wave finalize; cross-block reduction done with
//         V_WMMA_F32_16X16X4_F32 against an all-ones B matrix (f32-exact).
// Deterministic: fixed grid, fixed order, no float atomics.
// ---------------------------------------------------------------------------

#define NB  1024   // blocks in pass 1 (ws usage: 4*NB floats = 16 KB)
#define TPB 256    // 8 wave32s per block

typedef __attribute__((ext_vector_type(2))) float v2f;
typedef __attribute__((ext_vector_type(4))) float v4f;
typedef __attribute__((ext_vector_type(8))) float v8f;

__device__ __forceinline__ void cell_accum(float conf, float tgt,
                                           float d1, float d2, float d3, float d4,
                                           float& face, float& box, float& bce, float& bg)
{
    float m   = (tgt > 0.5f) ? 1.0f : 0.0f;
    float se  = d1 * d1 + d2 * d2 + d3 * d3 + d4 * d4;
    float lc  = fmaxf(logf(conf),    -100.0f);   // torch BCE clamp
    float l1c = fmaxf(log1pf(-conf), -100.0f);
    float b   = -(tgt * lc + (1.0f - tgt) * l1c);
    face += m;
    box  += m * se;
    bce  += m * b;
    bg   += (1.0f - m) * (-l1c);
}

__global__ __launch_bounds__(TPB) void mloss_partial(const float* __restrict__ x,
                                                     const float* __restrict__ y,
                                                     float* __restrict__ ws,
                                                     int nGroups, int nCells)
{
    float face = 0.0f, box = 0.0f, bce = 0.0f, bg = 0.0f;

    int tid    = blockIdx.x * blockDim.x + threadIdx.x;
    int stride = gridDim.x * blockDim.x;

    // Main loop: 4 cells (20 floats = five 16B-aligned vector loads) per iter.
    for (int g = tid; g < nGroups; g += stride) {
        size_t base = (size_t)g * 20u;
        const v4f* xp = (const v4f*)(x + base);
        const v4f* yp = (const v4f*)(y + base);
        v4f xv[5], yv[5];
#pragma unroll
        for (int i = 0; i < 5; ++i) {
            xv[i] = __builtin_nontemporal_load(xp + i);   // TH=NT: stream, don't pollute L2
            yv[i] = __builtin_nontemporal_load(yp + i);
        }
        const float* xs = (const float*)xv;
        const float* ys = (const float*)yv;
#pragma unroll
        for (int c = 0; c < 4; ++c) {
            cell_accum(xs[5 * c + 0], ys[5 * c + 0],
                       xs[5 * c + 1] - ys[5 * c + 1],
                       xs[5 * c + 2] - ys[5 * c + 2],
                       xs[5 * c + 3] - ys[5 * c + 3],
                       xs[5 * c + 4] - ys[5 * c + 4],
                       face, box, bce, bg);
        }
    }

    // Tail cells (nCells not multiple of 4): handled by thread 0 only.
    if (tid == 0) {
        for (int cIdx = nGroups * 4; cIdx < nCells; ++cIdx) {
            size_t base = (size_t)cIdx * 5u;
            cell_accum(x[base + 0], y[base + 0],
                       x[base + 1] - y[base + 1],
                       x[base + 2] - y[base + 2],
                       x[base + 3] - y[base + 3],
                       x[base + 4] - y[base + 4],
                       face, box, bce, bg);
        }
    }

    // wave32 butterfly reduce
#pragma unroll
    for (int off = 16; off > 0; off >>= 1) {
        face += __shfl_xor(face, off, 32);
        box  += __shfl_xor(box,  off, 32);
        bce  += __shfl_xor(bce,  off, 32);
        bg   += __shfl_xor(bg,   off, 32);
    }

    __shared__ float sdata[TPB / 32][4];
    int lane = threadIdx.x & 31;
    int wave = threadIdx.x >> 5;
    if (lane == 0) {
        sdata[wave][0] = face;
        sdata[wave][1] = box;
        sdata[wave][2] = bce;
        sdata[wave][3] = bg;
    }
    __syncthreads();

    // one partial per accumulator per block; SoA layout ws[a*NB + block]
    if (threadIdx.x < 4) {
        float s = 0.0f;
#pragma unroll
        for (int w = 0; w < TPB / 32; ++w) s += sdata[w][threadIdx.x];
        ws[threadIdx.x * NB + blockIdx.x] = s;
    }
}

// Finalize: one wave32. Cross-block reduction via V_WMMA_F32_16X16X4_F32.
// B = ones(4x16)  =>  D[m][n] = sum_k A[m][k] + C[m][n]  (row-sum, f32-exact path).
// Row m serves accumulator class (m & 3); per WMMA each class consumes 16 partials.
__global__ __launch_bounds__(32) void mloss_final(const float* __restrict__ ws,
                                                  float* __restrict__ out,
                                                  float totalCells)
{
    int lane = threadIdx.x;
    int m    = lane & 15;          // A-matrix row: lanes 0-15 -> K{0,1}, lanes 16-31 -> K{2,3}
    int kb   = (lane >> 4) << 1;   // 0 or 2
    int a    = m & 3;              // accumulator class of this row
    int g    = m >> 2;             // row-group within class

    v8f acc  = {0.0f, 0.0f, 0.0f, 0.0f, 0.0f, 0.0f, 0.0f, 0.0f};
    v2f bmat = {1.0f, 1.0f};       // all-ones B regardless of layout

    const float* p = ws + a * NB + g * 4 + kb;
    for (int i = 0; i < NB / 16; ++i) {
        v2f av;
        av.x = p[i * 16 + 0];
        av.y = p[i * 16 + 1];
        // D = A x ones + C   (8 args: neg_a, A, neg_b, B, c_mod, C, reuse_a, reuse_b)
        acc = __builtin_amdgcn_wmma_f32_16x16x4_f32(false, av, false, bmat,
                                                    (short)0, acc, false, false);
    }

    // C/D layout: vgpr r holds D[r][lane] (lanes 0-15) / D[r+8][lane-16] (lanes 16-31).
    // Class a lives in rows {a, a+4, a+8, a+12}: fold r and r+4, then lane0 + lane16.
    float s0 = acc[0] + acc[4];
    float s1 = acc[1] + acc[5];
    float s2 = acc[2] + acc[6];
    float s3 = acc[3] + acc[7];

    float face = __shfl(s0, 0, 32) + __shfl(s0, 16, 32);
    float box  = __shfl(s1, 0, 32) + __shfl(s1, 16, 32);
    float bce  = __shfl(s2, 0, 32) + __shfl(s2, 16, 32);
    float bg   = __shfl(s3, 0, 32) + __shfl(s3, 16, 32);

    if (lane == 0) {
        float bgNum    = totalCells - face;
        float diff_box = box / (4.0f * face);
        float diff_c   = bce / face;
        float diff_bg  = bg / bgNum;
        float scale    = 1.0f + 1.0f / face;
        out[0] = scale * diff_box + scale * diff_c + diff_bg;  // ALPHA = 1.0
    }
}

extern "C" void kernel_launch(void* const* d_in, const int* in_sizes, int n_in,
                              void* d_out, int out_size, void* d_ws, size_t ws_size,
                              hipStream_t stream)
{
    const float* x = (const float*)d_in[0];
    const float* y = (const float*)d_in[1];
    float* out = (float*)d_out;
    float* ws  = (float*)d_ws;   // needs 4*NB floats = 16 KB

    const int nCells  = in_sizes[0] / 5;   // B*C = 2048*2704
    const int nGroups = nCells / 4;        // 4 cells (80B) per thread-iteration

    mloss_partial<<<NB, TPB, 0, stream>>>(x, y, ws, nGroups, nCells);
    mloss_final<<<1, 32, 0, stream>>>(ws, out, (float)nCells);
}